// HierarchicalECPredictor_29515015258224
// MI455X (gfx1250) — compile-verified
//
#include <hip/hip_runtime.h>
#include <hip/hip_bf16.h>
#include <math.h>

// ---------------------------------------------------------------------------
// HierarchicalECPredictor for MI455X (gfx1250)
//   - segment-mean pooling: HBM-bound (1.02 GB stream), sorted-run register
//     accumulation + boundary float atomics, non-temporal B128 loads
//   - MLP chain: V_WMMA_F32_16X16X4_F32, compile-time-specialized GEMMs:
//     TN independent 16x16 tiles per wave (ILP across WMMA chains),
//     edge-free fast path when N tiles divide exactly, otherwise
//     branch-free clamp+select guards (EXEC stays all-1s)
// ---------------------------------------------------------------------------

typedef __attribute__((ext_vector_type(2))) float v2f;
typedef __attribute__((ext_vector_type(4))) float v4f;
typedef __attribute__((ext_vector_type(8))) float v8f;

#define G_GRAPHS 4096
#define H_DIM    256
#define M_ROWS   4096

__device__ __forceinline__ float gelu_exact(float x) {
    return 0.5f * x * (1.0f + erff(x * 0.70710678118654752440f));
}

// ---------------------------------------------------------------------------
__global__ __launch_bounds__(256) void zero_ws_kernel(float* __restrict__ p, int n) {
    int i = blockIdx.x * 256 + threadIdx.x;
    if (i < n) p[i] = 0.0f;
}

// ---------------------------------------------------------------------------
// Segment-sum pooling over sorted batch_idx (int64). Block = 256 threads:
// 4 groups x 64 lanes; group handles 64 consecutive nodes, lane c handles
// dims [4c,4c+3] (NT b128 loads, coalesced). Register run accumulation;
// atomics only at segment boundaries (~1.3 flushes / 64-node slab).
// ---------------------------------------------------------------------------
__global__ __launch_bounds__(256) void pool_kernel(const float* __restrict__ emb,
                                                   const long long* __restrict__ bidx,
                                                   float* __restrict__ sums,
                                                   float* __restrict__ counts,
                                                   int n) {
    const int g = threadIdx.x >> 6;
    const int c = threadIdx.x & 63;
    const int base = blockIdx.x * 256 + g * 64;

    float ax = 0.f, ay = 0.f, az = 0.f, aw = 0.f;
    int curSeg = -1;
    int run = 0;

    for (int i = 0; i < 64; ++i) {
        const int node = base + i;
        if (node >= n) break;
        const int seg = (int)bidx[node];
        if (seg != curSeg) {
            if (curSeg >= 0) {
                float* dst = sums + (size_t)curSeg * H_DIM + c * 4;
                atomicAdd(dst + 0, ax); atomicAdd(dst + 1, ay);
                atomicAdd(dst + 2, az); atomicAdd(dst + 3, aw);
                if (c == 0) atomicAdd(counts + curSeg, (float)run);
            }
            curSeg = seg; ax = ay = az = aw = 0.f; run = 0;
        }
        const v4f v = __builtin_nontemporal_load(
            (const v4f*)(emb + (size_t)node * H_DIM + c * 4));
        ax += v.x; ay += v.y; az += v.z; aw += v.w;
        ++run;
    }
    if (curSeg >= 0) {
        float* dst = sums + (size_t)curSeg * H_DIM + c * 4;
        atomicAdd(dst + 0, ax); atomicAdd(dst + 1, ay);
        atomicAdd(dst + 2, az); atomicAdd(dst + 3, aw);
        if (c == 0) atomicAdd(counts + curSeg, (float)run);
    }
}

__global__ __launch_bounds__(256) void inv_counts_kernel(const float* __restrict__ counts,
                                                         float* __restrict__ inv) {
    int i = blockIdx.x * 256 + threadIdx.x;
    if (i < G_GRAPHS) inv[i] = 1.0f / fmaxf(counts[i], 1.0f);
}

// ---------------------------------------------------------------------------
// Branch-free tail A element: A = [A1 (K1 cols) | A2 (K2 cols)], zero-padded.
// Clamped addresses -> unconditional loads, value selects (v_cndmask).
// ---------------------------------------------------------------------------
template <int K1, int K2>
__device__ __forceinline__ float tailA(const float* __restrict__ A1,
                                       const float* __restrict__ A2,
                                       int row, int t) {
    float v1 = 0.f, v2 = 0.f;
    if constexpr (K1 > 0) {
        int tt = t < K1 ? t : (K1 - 1);
        v1 = A1[(size_t)row * K1 + tt];
    }
    if constexpr (K2 > 0) {
        int u = t - K1;
        int uu = u < 0 ? 0 : (u < K2 ? u : (K2 - 1));
        v2 = A2[(size_t)row * K2 + uu];
    }
    return (t < K1) ? v1 : ((t < K1 + K2) ? v2 : 0.f);
}

// ---------------------------------------------------------------------------
// WMMA f32 GEMM: OUT[4096,N] = act(A[4096,K0+K1+K2] @ B + bias)
//   A = [A0 (K0, opt. per-row SCALE) | A1 (K1) | A2 (K2)], B row-major KxN.
// One wave computes TN 16x16 tiles (shared A fragment, TN independent WMMA
// accumulator chains). K0 % 4 == 0 -> guard-free main loop. When N's tile
// grid divides exactly (EDGE==false) all column guards vanish.
// Fragment layout per ISA 7.12.2 (32-bit, wave32):
//   A 16x4 : lane<16 -> row=lane, {V0,V1}={K0,K1}; lanes 16-31 -> {K2,K3}
//   B 4x16 : lane<16 -> col=lane, {V0,V1}={K0,K1}; lanes 16-31 -> {K2,K3}
//   C/D    : VGPR r -> row = m0 + r + 8*(lane>>4), col = n0 + (lane&15)
// ---------------------------------------------------------------------------
template <int K0, int K1, int K2, int N, int TN, bool SCALE, bool ACT>
__global__ __launch_bounds__(256) void gemm_wmma(
    const float* __restrict__ A0, const float* __restrict__ rowScale,
    const float* __restrict__ A1, const float* __restrict__ A2,
    const float* __restrict__ B, const float* __restrict__ bias,
    float* __restrict__ out) {

    constexpr int K = K0 + K1 + K2;
    constexpr int tilesN = (N + 15) / 16;
    constexpr int nGroups = (tilesN + TN - 1) / TN;
    constexpr bool EDGE = (N % 16 != 0) || (tilesN % TN != 0);

    const int waveId = blockIdx.x * 8 + (int)(threadIdx.x >> 5);
    if (waveId >= (M_ROWS / 16) * nGroups) return;   // wave-uniform

    const int tm  = waveId / nGroups;
    const int grp = waveId - tm * nGroups;
    const int m0  = tm << 4;

    const int lane = threadIdx.x & 31;
    const int half = lane >> 4;
    const int l16  = lane & 15;
    const int arow = m0 + l16;
    const float sc = SCALE ? rowScale[arow] : 1.0f;

    int colv[TN];    // true output column per tile
    int colc[TN];    // clamped column for loads (== colv when !EDGE)
    #pragma unroll
    for (int j = 0; j < TN; ++j) {
        colv[j] = (grp * TN + j) * 16 + l16;
        colc[j] = (EDGE && colv[j] >= N) ? (N - 1) : colv[j];
    }

    v8f acc[TN];
    #pragma unroll
    for (int j = 0; j < TN; ++j) acc[j] = (v8f){0.f,0.f,0.f,0.f,0.f,0.f,0.f,0.f};

    // ---- main loop over A0 (K0 % 4 == 0): guard-free A ----
    #pragma unroll 4
    for (int k = 0; k < K0; k += 4) {
        const int kk = k + 2 * half;                       // even -> 8B aligned
        v2f a = *(const v2f*)(A0 + (size_t)arow * K0 + kk);
        if constexpr (SCALE) { a.x *= sc; a.y *= sc; }
        #pragma unroll
        for (int j = 0; j < TN; ++j) {
            v2f b;
            float b0 = B[(size_t)kk * N + colc[j]];
            float b1 = B[(size_t)(kk + 1) * N + colc[j]];
            if constexpr (EDGE) {
                b.x = (colv[j] < N) ? b0 : 0.f;
                b.y = (colv[j] < N) ? b1 : 0.f;
            } else {
                b.x = b0; b.y = b1;
            }
            acc[j] = __builtin_amdgcn_wmma_f32_16x16x4_f32(
                false, a, false, b, (short)0, acc[j], false, false);
        }
    }

    // ---- tail over concatenated A1|A2 (zero-padded to multiple of 4) ----
    if constexpr (K1 + K2 > 0) {
        constexpr int KTP = (K1 + K2 + 3) & ~3;
        #pragma unroll
        for (int k = 0; k < KTP; k += 4) {
            const int kk = k + 2 * half;
            v2f a;
            a.x = tailA<K1, K2>(A1, A2, arow, kk);
            a.y = tailA<K1, K2>(A1, A2, arow, kk + 1);
            const int kg0 = K0 + kk;
            const int kc0 = kg0 < K ? kg0 : (K - 1);
            const int kc1 = (kg0 + 1) < K ? (kg0 + 1) : (K - 1);
            #pragma unroll
            for (int j = 0; j < TN; ++j) {
                float b0 = B[(size_t)kc0 * N + colc[j]];
                float b1 = B[(size_t)kc1 * N + colc[j]];
                const bool colOK = !EDGE || (colv[j] < N);
                v2f b;
                b.x = (kg0     < K && colOK) ? b0 : 0.f;
                b.y = (kg0 + 1 < K && colOK) ? b1 : 0.f;
                acc[j] = __builtin_amdgcn_wmma_f32_16x16x4_f32(
                    false, a, false, b, (short)0, acc[j], false, false);
            }
        }
    }

    // ---- epilogue: bias + optional exact GELU, masked store only on edge ----
    #pragma unroll
    for (int j = 0; j < TN; ++j) {
        if (!EDGE || colv[j] < N) {
            const float bv = bias[colv[j]];
            #pragma unroll
            for (int r = 0; r < 8; ++r) {
                const int row = m0 + r + 8 * half;
                float v = acc[j][r] + bv;
                if constexpr (ACT) v = gelu_exact(v);
                out[(size_t)row * N + colv[j]] = v;
            }
        }
    }
}

// ---------------------------------------------------------------------------
__global__ __launch_bounds__(256) void softmax_rows(const float* __restrict__ x,
                                                    float* __restrict__ p,
                                                    int M, int C) {
    int r = blockIdx.x * 256 + threadIdx.x;
    if (r >= M) return;
    const float* xr = x + (size_t)r * C;
    float* pr = p + (size_t)r * C;
    float m = -INFINITY;
    for (int j = 0; j < C; ++j) m = fmaxf(m, xr[j]);
    float s = 0.f;
    for (int j = 0; j < C; ++j) { float e = expf(xr[j] - m); pr[j] = e; s += e; }
    const float is = 1.0f / s;
    for (int j = 0; j < C; ++j) pr[j] *= is;
}

// ---------------------------------------------------------------------------
static inline int gemmBlocks(int N, int TN) {
    int tilesN = (N + 15) / 16;
    int nGroups = (tilesN + TN - 1) / TN;
    int waves = (M_ROWS / 16) * nGroups;
    return (waves + 7) / 8;           // 8 waves (256 threads) per block
}

extern "C" void kernel_launch(void* const* d_in, const int* in_sizes, int n_in,
                              void* d_out, int out_size, void* d_ws, size_t ws_size,
                              hipStream_t stream) {
    const float*     emb  = (const float*)d_in[0];
    const long long* bidx = (const long long*)d_in[1];     // jnp.int64
    const float* Wp  = (const float*)d_in[2];
    const float* bp  = (const float*)d_in[3];
    const float* W1a = (const float*)d_in[4];
    const float* b1a = (const float*)d_in[5];
    const float* W1b = (const float*)d_in[6];
    const float* b1b = (const float*)d_in[7];
    const float* W2a = (const float*)d_in[8];
    const float* b2a = (const float*)d_in[9];
    const float* W2b = (const float*)d_in[10];
    const float* b2b = (const float*)d_in[11];
    const float* W3a = (const float*)d_in[12];
    const float* b3a = (const float*)d_in[13];
    const float* W3b = (const float*)d_in[14];
    const float* b3b = (const float*)d_in[15];

    const int nNodes = in_sizes[1];

    // workspace (floats); hidden/probs reuse the dead `sums` region
    float* ws     = (float*)d_ws;
    float* sums   = ws;                                   // 4096*256
    float* counts = ws + (size_t)G_GRAPHS * H_DIM;        // 4096
    float* inv    = counts + G_GRAPHS;                    // 4096
    float* hidden = sums;                                 // 4096*128 (after stage B)
    float* probs1 = sums + (size_t)G_GRAPHS * 128;        // 4096*7
    float* probs2 = probs1 + (size_t)G_GRAPHS * 7;        // 4096*70

    // outputs, concatenated flat in return order
    float* out = (float*)d_out;
    float* ec1 = out;                                     // (G,7)
    float* ec2 = ec1 + (size_t)G_GRAPHS * 7;              // (G,70)
    float* ec3 = ec2 + (size_t)G_GRAPHS * 70;             // (G,300)
    float* gf  = ec3 + (size_t)G_GRAPHS * 300;            // (G,256)

    // 1) zero accumulators
    const int zn = G_GRAPHS * H_DIM + 2 * G_GRAPHS;
    zero_ws_kernel<<<(zn + 255) / 256, 256, 0, stream>>>(ws, zn);

    // 2) segment sums + counts (the ~44us HBM-bound pass)
    pool_kernel<<<(nNodes + 255) / 256, 256, 0, stream>>>(emb, bidx, sums, counts, nNodes);

    // 3) 1/max(count,1), fused into stage B as per-row scale (== mean)
    inv_counts_kernel<<<(G_GRAPHS + 255) / 256, 256, 0, stream>>>(counts, inv);

    // 4) global_feat = gelu((sums * inv) @ Wp + bp)           K=256, N=256
    gemm_wmma<256,0,0,256,4,true,true><<<gemmBlocks(256,4), 256, 0, stream>>>(
        sums, inv, nullptr, nullptr, Wp, bp, gf);

    // 5) ec1 head                                             K=256->128->7
    gemm_wmma<256,0,0,128,4,false,true><<<gemmBlocks(128,4), 256, 0, stream>>>(
        gf, nullptr, nullptr, nullptr, W1a, b1a, hidden);
    gemm_wmma<128,0,0,7,1,false,false><<<gemmBlocks(7,1), 256, 0, stream>>>(
        hidden, nullptr, nullptr, nullptr, W1b, b1b, ec1);
    softmax_rows<<<(M_ROWS + 255) / 256, 256, 0, stream>>>(ec1, probs1, M_ROWS, 7);

    // 6) ec2 head: A = [gf | probs1]                          K=263->128->70
    gemm_wmma<256,7,0,128,4,false,true><<<gemmBlocks(128,4), 256, 0, stream>>>(
        gf, nullptr, probs1, nullptr, W2a, b2a, hidden);
    gemm_wmma<128,0,0,70,5,false,false><<<gemmBlocks(70,5), 256, 0, stream>>>(
        hidden, nullptr, nullptr, nullptr, W2b, b2b, ec2);
    softmax_rows<<<(M_ROWS + 255) / 256, 256, 0, stream>>>(ec2, probs2, M_ROWS, 70);

    // 7) ec3 head: A = [gf | probs1 | probs2]                 K=333->128->300
    gemm_wmma<256,7,70,128,4,false,true><<<gemmBlocks(128,4), 256, 0, stream>>>(
        gf, nullptr, probs1, probs2, W3a, b3a, hidden);
    gemm_wmma<128,0,0,300,4,false,false><<<gemmBlocks(300,4), 256, 0, stream>>>(
        hidden, nullptr, nullptr, nullptr, W3b, b3b, ec3);
}